// QuantumCircuitLayer_38955353375497
// MI455X (gfx1250) — compile-verified
//
#include <hip/hip_runtime.h>

// Quantum circuit sim, 12 qubits, batch 2048.
// One sample per 256-thread block; 4096-complex state lives in LDS.
// Per layer, the 12 fused (Rz*Ry*Rx) gates are combined into three 16x16
// complex matrices (Kronecker over 4-qubit groups) and applied as
// 16 complex 16x16x16 GEMMs per group via V_WMMA_F32_16X16X4_F32
// (4 chained WMMAs over K per real GEMM, 4 real GEMMs per complex GEMM).
// Each 16-wide tile's read and write footprints coincide and are owned by a
// single wave -> in-place, single buffer, one barrier per pass.
// CNOT ring = index permutation pass. Expectations = deterministic reduction.

#define NQ      12
#define DIM     4096
#define NLAYERS 2
#define NMAT    6            // 2 layers * 3 groups
#define PI_F    3.14159265358979323846f

typedef __attribute__((ext_vector_type(2))) float v2f;
typedef __attribute__((ext_vector_type(8))) float v8f;

__device__ __forceinline__ v8f wmma4(v2f a, v2f b, v8f c) {
  // D(16x16) = A(16x4) x B(4x16) + C, fp32
  return __builtin_amdgcn_wmma_f32_16x16x4_f32(false, a, false, b,
                                               (short)0, c, false, false);
}

// ---------------------------------------------------------------------------
// Prep: build the 6 group matrices Ug = kron(U_{4g},U_{4g+1},U_{4g+2},U_{4g+3})
// with U = Rz(t3)*Ry(t2)*Rx(t1) per qubit. 1 block, 256 threads: thread t
// computes entry (r=t>>4, k=t&15) of each of the 6 matrices.
// ---------------------------------------------------------------------------
__global__ void prep_gates(const float* __restrict__ wt,
                           float* __restrict__ Ur, float* __restrict__ Ui) {
  int t = threadIdx.x;
  int r = t >> 4, k = t & 15;
  for (int l = 0; l < NLAYERS; ++l) {
    for (int g = 0; g < 3; ++g) {
      float pr = 1.f, pi = 0.f;
      #pragma unroll
      for (int j = 0; j < 4; ++j) {
        int wire = g * 4 + j;
        int base = (l * NQ + wire) * 3;
        float tx = wt[base + 0] * 0.5f;
        float ty = wt[base + 1] * 0.5f;
        float tz = wt[base + 2] * 0.5f;
        float cx = cosf(tx), sx = sinf(tx);
        float cy = cosf(ty), sy = sinf(ty);
        float cz = cosf(tz), sz = sinf(tz);
        // M = Ry*Rx:
        // M00 =  cy*cx + i*sy*sx   M01 = -sy*cx - i*cy*sx
        // M10 =  sy*cx - i*cy*sx   M11 =  cy*cx - i*sy*sx
        // U = Rz*M: row0 *= (cz - i*sz), row1 *= (cz + i*sz)
        int rb = (r >> (3 - j)) & 1;   // wire 4g+j <-> group-local bit 3-j
        int kb = (k >> (3 - j)) & 1;
        float mr, mi;
        if (!rb && !kb)      { mr =  cy * cx; mi =  sy * sx; }
        else if (!rb &&  kb) { mr = -sy * cx; mi = -cy * sx; }
        else if ( rb && !kb) { mr =  sy * cx; mi = -cy * sx; }
        else                 { mr =  cy * cx; mi = -sy * sx; }
        float er = cz, ei = rb ? sz : -sz;
        float ur = mr * er - mi * ei;
        float ui = mr * ei + mi * er;
        float nr = pr * ur - pi * ui;
        float ni = pr * ui + pi * ur;
        pr = nr; pi = ni;
      }
      int m = l * 3 + g;
      Ur[m * 256 + r * 16 + k] = pr;
      Ui[m * 256 + r * 16 + k] = pi;
    }
  }
}

// ---------------------------------------------------------------------------
// One group contraction: apply 16x16 complex Ug to the 4-bit slice of the
// state. Group 0 -> bits 11..8, group 1 -> bits 7..4, group 2 -> bits 3..0.
// 16 tiles; wave w owns tiles 2w, 2w+1. In-place.
// WMMA fp32 fragment layouts (ISA 7.12.2):
//   A 16x4 : lane = M (0..15, twice); VGPR0/1 = K+khalf, K+khalf+1 (khalf=0|2)
//   B 4x16 : lane = N (mirrored);     VGPR0/1 = K+khalf, K+khalf+1
//   C/D    : VGPR j, lanes 0-15 -> (M=j,   N=lane)
//                    lanes16-31 -> (M=j+8, N=lane-16)
// ---------------------------------------------------------------------------
template <int G>
__device__ __forceinline__ void group_pass(float* sre, float* simg,
                                           const float* Ur, const float* Ui,
                                           int lane, int wave) {
  int laneM = lane & 15;
  int khalf = (lane >> 4) << 1;  // 0 or 2
  int mhi   = (lane >> 4) << 3;  // D rows: +0 or +8
  #pragma unroll
  for (int tt = 0; tt < 2; ++tt) {
    int t = wave * 2 + tt;
    v2f Ar[4], Aim[4], Br[4], Bi[4];
    #pragma unroll
    for (int kc = 0; kc < 4; ++kc) {
      int K = kc * 4 + khalf;
      if (G == 0 || G == 1) {             // A = Ug (row-major [M][K])
        int ai = laneM * 16 + K;
        Ar[kc][0] = Ur[ai];  Ar[kc][1] = Ur[ai + 1];
        Aim[kc][0] = Ui[ai]; Aim[kc][1] = Ui[ai + 1];
      } else {                            // A = state rows: t*256 + m*16 + K
        int ai = t * 256 + laneM * 16 + K;
        Ar[kc][0] = sre[ai];  Ar[kc][1] = sre[ai + 1];
        Aim[kc][0] = simg[ai]; Aim[kc][1] = simg[ai + 1];
      }
      if (G == 0) {                       // B[K,n] = s[K*256 + t*16 + n]
        int b0 = K * 256 + t * 16 + laneM;
        Br[kc][0] = sre[b0];  Br[kc][1] = sre[b0 + 256];
        Bi[kc][0] = simg[b0]; Bi[kc][1] = simg[b0 + 256];
      } else if (G == 1) {                // B[K,n] = s[t*256 + K*16 + n]
        int b0 = t * 256 + K * 16 + laneM;
        Br[kc][0] = sre[b0];  Br[kc][1] = sre[b0 + 16];
        Bi[kc][0] = simg[b0]; Bi[kc][1] = simg[b0 + 16];
      } else {                            // B = Ug^T: B[K,n] = Ug[n*16 + K]
        int b0 = laneM * 16 + K;
        Br[kc][0] = Ur[b0];  Br[kc][1] = Ur[b0 + 1];
        Bi[kc][0] = Ui[b0];  Bi[kc][1] = Ui[b0 + 1];
      }
    }
    v8f Dr = {0, 0, 0, 0, 0, 0, 0, 0};
    v8f Di = {0, 0, 0, 0, 0, 0, 0, 0};
    #pragma unroll
    for (int kc = 0; kc < 4; ++kc) Dr = wmma4(Ar[kc], Br[kc], Dr);
    #pragma unroll
    for (int kc = 0; kc < 4; ++kc) {      // f32 WMMA has no A-neg: negate in VALU
      v2f an = -Aim[kc];
      Dr = wmma4(an, Bi[kc], Dr);
    }
    #pragma unroll
    for (int kc = 0; kc < 4; ++kc) Di = wmma4(Ar[kc], Bi[kc], Di);
    #pragma unroll
    for (int kc = 0; kc < 4; ++kc) Di = wmma4(Aim[kc], Br[kc], Di);
    #pragma unroll
    for (int j = 0; j < 8; ++j) {
      int M = j + mhi;
      int di = (G == 0) ? (M * 256 + t * 16 + laneM)
                        : (t * 256 + M * 16 + laneM);
      sre[di] = Dr[j];
      simg[di] = Di[j];
    }
  }
}

// CNOT ring permutation: s_final[i] = s[p01(p12(...p_{10,11}(p_{11,0}(i))...))]
// wire w <-> bit (11-w); p_{c,t}: flip target bit if control bit set.
__device__ __forceinline__ int cnot_src(int i) {
  int j = i;
  j ^= (j & 1) << 11;                    // CNOT(11,0): ctrl bit0, tgt bit11
  #pragma unroll
  for (int q = 10; q >= 0; --q) {
    int cb = 11 - q, tb = 10 - q;        // CNOT(q, q+1)
    j ^= ((j >> cb) & 1) << tb;
  }
  return j;
}

__global__ __launch_bounds__(256) void sim_kernel(const float* __restrict__ x,
                                                  const float* __restrict__ gUr,
                                                  const float* __restrict__ gUi,
                                                  float* __restrict__ out) {
  __shared__ float sre[DIM];
  __shared__ float simg[DIM];
  __shared__ float Ugr[NMAT * 256];
  __shared__ float Ugi[NMAT * 256];
  __shared__ float csA[NQ], snA[NQ];
  __shared__ float redW[8][NQ];

  const int tid  = threadIdx.x;
  const int b    = blockIdx.x;
  const int lane = tid & 31;
  const int wave = tid >> 5;

  for (int i = tid; i < NMAT * 256; i += 256) { Ugr[i] = gUr[i]; Ugi[i] = gUi[i]; }
  if (tid < NQ) {
    float half = x[b * NQ + tid] * (PI_F * 0.5f);
    csA[tid] = cosf(half);
    snA[tid] = sinf(half);
  }
  __syncthreads();

  // Initial separable real product state (RY layer on |0..0>)
  {
    float base = 1.f;
    #pragma unroll
    for (int w = 0; w < 8; ++w) {        // wires 0..7 live in bits 11..4 = tid[7:0]
      int bit = (tid >> (7 - w)) & 1;
      base *= bit ? snA[w] : csA[w];
    }
    #pragma unroll
    for (int k = 0; k < 16; ++k) {
      float a = base;
      #pragma unroll
      for (int w = 8; w < 12; ++w) {     // wires 8..11 live in bits 3..0 = k
        int bit = (k >> (11 - w)) & 1;
        a *= bit ? snA[w] : csA[w];
      }
      sre[tid * 16 + k] = a;
      simg[tid * 16 + k] = 0.f;
    }
  }

  for (int l = 0; l < NLAYERS; ++l) {
    __syncthreads();
    group_pass<0>(sre, simg, Ugr + (l * 3 + 0) * 256, Ugi + (l * 3 + 0) * 256, lane, wave);
    __syncthreads();
    group_pass<1>(sre, simg, Ugr + (l * 3 + 1) * 256, Ugi + (l * 3 + 1) * 256, lane, wave);
    __syncthreads();
    group_pass<2>(sre, simg, Ugr + (l * 3 + 2) * 256, Ugi + (l * 3 + 2) * 256, lane, wave);
    __syncthreads();
    // CNOT ring: read sources to regs, barrier, write in place
    float tr[16], ti[16];
    #pragma unroll
    for (int k = 0; k < 16; ++k) {
      int j = cnot_src(tid * 16 + k);
      tr[k] = sre[j];
      ti[k] = simg[j];
    }
    __syncthreads();
    #pragma unroll
    for (int k = 0; k < 16; ++k) {
      sre[tid * 16 + k] = tr[k];
      simg[tid * 16 + k] = ti[k];
    }
  }
  __syncthreads();

  // out[b,w] = sum_i |amp_i|^2 * (1 - 2*bit_w(i))
  float acc[NQ];
  #pragma unroll
  for (int w = 0; w < NQ; ++w) acc[w] = 0.f;
  #pragma unroll
  for (int k = 0; k < 16; ++k) {
    int i = tid * 16 + k;
    float re = sre[i], im = simg[i];
    float p = re * re + im * im;
    #pragma unroll
    for (int w = 0; w < NQ; ++w)
      acc[w] += ((i >> (11 - w)) & 1) ? -p : p;
  }
  #pragma unroll
  for (int w = 0; w < NQ; ++w) {
    float v = acc[w];
    #pragma unroll
    for (int m = 16; m >= 1; m >>= 1) v += __shfl_xor(v, m, 32);
    if (lane == 0) redW[wave][w] = v;
  }
  __syncthreads();
  if (tid < NQ) {
    float s = 0.f;                       // fixed-order sum -> deterministic
    #pragma unroll
    for (int wv = 0; wv < 8; ++wv) s += redW[wv][tid];
    out[b * NQ + tid] = s;
  }
}

extern "C" void kernel_launch(void* const* d_in, const int* in_sizes, int n_in,
                              void* d_out, int out_size, void* d_ws, size_t ws_size,
                              hipStream_t stream) {
  const float* x   = (const float*)d_in[0];
  const float* wts = (const float*)d_in[1];
  float* out = (float*)d_out;
  float* Ur = (float*)d_ws;              // 6*256 floats
  float* Ui = Ur + NMAT * 256;           // 6*256 floats (12 KB total)
  int B = in_sizes[0] / NQ;

  prep_gates<<<1, 256, 0, stream>>>(wts, Ur, Ui);
  sim_kernel<<<B, 256, 0, stream>>>(x, Ur, Ui, out);
}